// historicalNN_52192442581336
// MI455X (gfx1250) — compile-verified
//
#include <hip/hip_runtime.h>
#include <math.h>

typedef __attribute__((ext_vector_type(2))) float v2f;
typedef __attribute__((ext_vector_type(8))) float v8f;

#define NT     3684          // N*T = 307*12
#define NTILE  231           // ceil(NT/16)
#define NTPAD  (NTILE * 16)  // 3696
#define BSTRIDE (12 * 307 * 64)

// ---------- K1: tiny precompute of M = Wk^T Wq, kb = Wk^T bq, wbkq = bk^T Wq, bk.bq
__global__ void prep_kernel(const float* __restrict__ Wq, const float* __restrict__ bq,
                            const float* __restrict__ Wk, const float* __restrict__ bk,
                            float* __restrict__ Mm, float* __restrict__ kb,
                            float* __restrict__ wbkq, float* __restrict__ scal) {
    int tid = threadIdx.x;
    for (int e = tid; e < 4096; e += 256) {
        int cp = e >> 6, c = e & 63;
        float s = 0.f;
        for (int o = 0; o < 64; ++o) s += Wk[o * 64 + cp] * Wq[o * 64 + c];
        Mm[e] = s;
    }
    if (tid < 64) {
        float s1 = 0.f, s2 = 0.f;
        for (int o = 0; o < 64; ++o) {
            s1 += Wk[o * 64 + tid] * bq[o];
            s2 += bk[o] * Wq[o * 64 + tid];
        }
        kb[tid] = s1;
        wbkq[tid] = s2;
    }
    if (tid == 0) {
        float s = 0.f;
        for (int o = 0; o < 64; ++o) s += bk[o] * bq[o];
        scal[0] = s;
    }
}

// ---------- K2: xs[nt,c] = sum_b x[b,t,n,c]; per-block partial of sum(xs*wbkq)
__global__ void bsum_kernel(const float* __restrict__ x, const float* __restrict__ wbkq,
                            float* __restrict__ xs, float* __restrict__ bparts) {
    int idx = blockIdx.x * 256 + threadIdx.x;   // < NTPAD*64
    float partial = 0.f;
    int nt = idx >> 6, c = idx & 63;
    float s = 0.f;
    if (nt < NT) {
        int n = nt / 12, t = nt % 12;
        int base = (t * 307 + n) * 64 + c;      // x[b][t][n][c]
        s = x[base] + x[base + BSTRIDE] + x[base + 2 * BSTRIDE] + x[base + 3 * BSTRIDE];
        partial = s * wbkq[c];
    }
    xs[idx] = s;                                 // zero in padded region
    __shared__ float red[256];
    red[threadIdx.x] = partial;
    __syncthreads();
    for (int off = 128; off > 0; off >>= 1) {
        if (threadIdx.x < off) red[threadIdx.x] += red[threadIdx.x + off];
        __syncthreads();
    }
    if (threadIdx.x == 0) bparts[blockIdx.x] = red[0];
}

// ---------- K3: deterministic second-pass reduction for Sb partial
__global__ void sbsum_kernel(const float* __restrict__ bparts, float* __restrict__ scal) {
    __shared__ float red[256];
    float a = 0.f;
    for (int i = threadIdx.x; i < 924; i += 256) a += bparts[i];
    red[threadIdx.x] = a;
    __syncthreads();
    for (int off = 128; off > 0; off >>= 1) {
        if (threadIdx.x < off) red[threadIdx.x] += red[threadIdx.x + off];
        __syncthreads();
    }
    if (threadIdx.x == 0) scal[1] = red[0];
}

// ---------- shared WMMA f32 16x16x4 tile GEMM:
// D[16x16] = A[rows tile_i*16.., all 64 k] * B, with B[k][n] = Brow[n*64 + k]
// A layout: lane l, reg r -> A[M = l%16, K = 2*(l/16)+r]  (per k-step of 4)
// B layout: lane l, reg r -> B[K = 2*(l/16)+r, N = l%16]
__device__ inline v8f wmma_tile_64(const float* __restrict__ A, const float* __restrict__ Brow,
                                   int tile_i, int tile_j, int lane) {
    v8f acc = {};
    const int mrow = tile_i * 16 + (lane & 15);
    const int ncol = tile_j * 16 + (lane & 15);
    const int khalf = (lane >> 4) << 1;   // 0 or 2
    for (int k0 = 0; k0 < 64; k0 += 4) {
        v2f a, b;
        a.x = A[mrow * 64 + k0 + khalf];
        a.y = A[mrow * 64 + k0 + khalf + 1];
        const float* bp = &Brow[ncol * 64 + k0 + khalf];
        b.x = bp[0];
        b.y = bp[1];
        acc = __builtin_amdgcn_wmma_f32_16x16x4_f32(false, a, false, b,
                                                    (short)0, acc, false, false);
    }
    return acc;
}

// ---------- K4: QW[c', nt] = sum_c M[c',c]*xs[nt,c] + 4*kb[c']   (64 x NTPAD x 64 GEMM)
__global__ void qw_gemm_kernel(const float* __restrict__ Mm, const float* __restrict__ xs,
                               const float* __restrict__ kb, float* __restrict__ QW) {
    int wave = (blockIdx.x * blockDim.x + threadIdx.x) >> 5;
    int lane = threadIdx.x & 31;
    if (wave >= 4 * NTILE) return;              // wave-uniform
    int tile_i = wave / NTILE;
    int tile_j = wave % NTILE;
    v8f acc = wmma_tile_64(Mm, xs, tile_i, tile_j, lane);
    int ncol = tile_j * 16 + (lane & 15);
    int rbase = tile_i * 16 + ((lane >> 4) << 3);
#pragma unroll
    for (int r = 0; r < 8; ++r)
        QW[(rbase + r) * NTPAD + ncol] = acc[r] + 4.0f * kb[rbase + r];
}

// ---------- K5: sims_all[w] = sum_{t,n,c} QW[c, n*12+t] * hist[(30w+t)*307+n][c] + Sb
__global__ void sims_kernel(const float* __restrict__ QW, const float* __restrict__ hist,
                            const float* __restrict__ scal, float* __restrict__ sims) {
    const int w = blockIdx.x;                   // 0..199
    const int start = 30 * w;
    float acc = 0.f;
    const int total = 307 * 12 * 64;
    for (int i = threadIdx.x; i < total; i += 256) {
        int t = i / (307 * 64);
        int rem = i - t * (307 * 64);
        int n = rem >> 6;
        int c = rem & 63;
        acc += QW[c * NTPAD + n * 12 + t] * hist[((start + t) * 307 + n) * 64 + c];
    }
    __shared__ float red[256];
    red[threadIdx.x] = acc;
    __syncthreads();
    for (int off = 128; off > 0; off >>= 1) {
        if (threadIdx.x < off) red[threadIdx.x] += red[threadIdx.x + off];
        __syncthreads();
    }
    if (threadIdx.x == 0)
        sims[w] = red[0] + scal[1] + 14736.0f * scal[0];   // B*N*T = 4*307*12
}

// ---------- K6: top-10 of sims_all[1:]/numel + softmax (serial; matches JAX tie-break)
__global__ void topk_kernel(const float* __restrict__ sims, float* __restrict__ corr,
                            int* __restrict__ topidx) {
    if (threadIdx.x != 0) return;
    const float inv_numel = 1.0f / 943104.0f;   // 4*64*307*12
    float s[199];
    for (int i = 0; i < 199; ++i) s[i] = sims[i + 1] * inv_numel;
    float w[10]; int id[10];
    for (int k = 0; k < 10; ++k) {
        float best = -INFINITY; int bi = 0;
        for (int i = 0; i < 199; ++i)
            if (s[i] > best) { best = s[i]; bi = i; }
        w[k] = best; id[k] = bi; s[bi] = -INFINITY;
    }
    float mx = w[0];                            // sorted descending: w[0] is max
    float e[10], sum = 0.f;
    for (int k = 0; k < 10; ++k) { e[k] = __expf(w[k] - mx); sum += e[k]; }
    float inv = 1.0f / sum;
    for (int k = 0; k < 10; ++k) { corr[k] = e[k] * inv; topidx[k] = id[k]; }
}

// ---------- K7: H[nt,c] = sum_k corr[k] * hist[(30*idx_k + t)*307 + n][c]
// (reference indexes windows[index] with sims-space indices -> start = 30*idx_k)
__global__ void hgather_kernel(const float* __restrict__ hist, const float* __restrict__ corr,
                               const int* __restrict__ topidx, float* __restrict__ H) {
    int idx = blockIdx.x * 256 + threadIdx.x;   // < NTPAD*64
    int nt = idx >> 6, c = idx & 63;
    float h = 0.f;
    if (nt < NT) {
        int n = nt / 12, t = nt % 12;
#pragma unroll
        for (int k = 0; k < 10; ++k)
            h += corr[k] * hist[((30 * topidx[k] + t) * 307 + n) * 64 + c];
    }
    H[idx] = h;
}

// ---------- K8: agg[o,nt] = sum_c Wv[o,c]*H[nt,c] + bv[o]; out[b,t,n,o] = agg
__global__ void out_gemm_kernel(const float* __restrict__ Wv, const float* __restrict__ H,
                                const float* __restrict__ bv, float* __restrict__ out) {
    int wave = (blockIdx.x * blockDim.x + threadIdx.x) >> 5;
    int lane = threadIdx.x & 31;
    if (wave >= 4 * NTILE) return;              // wave-uniform
    int tile_i = wave / NTILE;
    int tile_j = wave % NTILE;
    v8f acc = wmma_tile_64(Wv, H, tile_i, tile_j, lane);
    int nt = tile_j * 16 + (lane & 15);
    if (nt >= NT) return;                       // after all WMMAs
    int n = nt / 12, t = nt % 12;
    int obase = tile_i * 16 + ((lane >> 4) << 3);
    float4 lo = make_float4(acc[0] + bv[obase + 0], acc[1] + bv[obase + 1],
                            acc[2] + bv[obase + 2], acc[3] + bv[obase + 3]);
    float4 hi = make_float4(acc[4] + bv[obase + 4], acc[5] + bv[obase + 5],
                            acc[6] + bv[obase + 6], acc[7] + bv[obase + 7]);
#pragma unroll
    for (int b = 0; b < 4; ++b) {
        float* p = out + (((b * 12 + t) * 307 + n) * 64) + obase;
        *(float4*)(p)     = lo;
        *(float4*)(p + 4) = hi;
    }
}

extern "C" void kernel_launch(void* const* d_in, const int* in_sizes, int n_in,
                              void* d_out, int out_size, void* d_ws, size_t ws_size,
                              hipStream_t stream) {
    (void)in_sizes; (void)n_in; (void)out_size; (void)ws_size;
    const float* x    = (const float*)d_in[0];
    const float* hist = (const float*)d_in[1];
    const float* Wq   = (const float*)d_in[2];
    const float* bq   = (const float*)d_in[3];
    const float* Wk   = (const float*)d_in[4];
    const float* bk   = (const float*)d_in[5];
    const float* Wv   = (const float*)d_in[6];
    const float* bv   = (const float*)d_in[7];
    float* out = (float*)d_out;

    float* ws     = (float*)d_ws;
    float* xs     = ws;                     // NTPAD*64
    float* QW     = xs + NTPAD * 64;        // 64*NTPAD
    float* H      = QW + 64 * NTPAD;        // NTPAD*64
    float* Mm     = H + NTPAD * 64;         // 4096
    float* kb     = Mm + 4096;              // 64
    float* wbkq   = kb + 64;                // 64
    float* scal   = wbkq + 64;              // 2: [0]=bk.bq, [1]=sum(xs*wbkq)
    float* sims   = scal + 2;               // 200
    float* corr   = sims + 200;             // 10
    int*   topidx = (int*)(corr + 10);      // 10
    float* bparts = (float*)(topidx + 10);  // 924

    const int elem_blocks = (NTPAD * 64) / 256;          // 924
    const int gemm_blocks = (4 * NTILE + 7) / 8;         // 924 waves / 8 per block

    prep_kernel<<<1, 256, 0, stream>>>(Wq, bq, Wk, bk, Mm, kb, wbkq, scal);
    bsum_kernel<<<elem_blocks, 256, 0, stream>>>(x, wbkq, xs, bparts);
    sbsum_kernel<<<1, 256, 0, stream>>>(bparts, scal);
    qw_gemm_kernel<<<gemm_blocks, 256, 0, stream>>>(Mm, xs, kb, QW);
    sims_kernel<<<200, 256, 0, stream>>>(QW, hist, scal, sims);
    topk_kernel<<<1, 64, 0, stream>>>(sims, corr, topidx);
    hgather_kernel<<<elem_blocks, 256, 0, stream>>>(hist, corr, topidx, H);
    out_gemm_kernel<<<gemm_blocks, 256, 0, stream>>>(Wv, H, bv, out);
}